// Decoder_15522011808493
// MI455X (gfx1250) — compile-verified
//
#include <hip/hip_runtime.h>
#include <hip/hip_bf16.h>

#define BS    128
#define LSEQ  512
#define ENC   512
#define DEC   1024
#define ATTN  512
#define EMB   256
#define VOCAB 128
#define TMAX  128

typedef __attribute__((ext_vector_type(2))) float v2f;
typedef __attribute__((ext_vector_type(8))) float v8f;

// ---------------------------------------------------------------------------
// WMMA f32 16x16x4 tile helpers (wave32).
// A fragment (16x4, MxK): lane = M%16; VGPR v holds K = (lane/16)*2 + v.
// B fragment (4x16, KxN): lane = N;   VGPR v holds K = (lane/16)*2 + v.
// C/D (16x16):            lane = N;   VGPR v holds M = 8*(lane/16) + v.
// BT=0: B row-major [K x N], row stride ldb.
// BT=1: B is weight matrix W[N x K] row-major (computes A * W^T).
// NT:   number of 16-wide N tiles per wave (A fragment reused NT times).
// ---------------------------------------------------------------------------
template <int BT, int NT>
__device__ __forceinline__ void wmma_accum_nt(const float* __restrict__ A, int lda,
                                              const float* __restrict__ B, int ldb,
                                              int K, int m0, int n0, v8f* acc) {
  const int lane = threadIdx.x & 31;
  const int half = lane >> 4;   // 0 or 1
  const int r    = lane & 15;   // M for A, N for B
  const float* ap = A + (size_t)(m0 + r) * lda;
  for (int k = 0; k < K; k += 4) {
    const int ka = k + half * 2;
    v2f a;
    a.x = ap[ka];
    a.y = ap[ka + 1];
#pragma unroll
    for (int j = 0; j < NT; ++j) {
      const int n = n0 + j * 16 + r;
      v2f b;
      if (BT == 0) {
        b.x = B[(size_t)ka * ldb + n];
        b.y = B[(size_t)(ka + 1) * ldb + n];
      } else {
        const float* bp = B + (size_t)n * ldb;
        b.x = bp[ka];
        b.y = bp[ka + 1];
      }
      acc[j] = __builtin_amdgcn_wmma_f32_16x16x4_f32(false, a, false, b,
                                                     (short)0, acc[j], false, false);
    }
  }
}

template <bool HAS_BN, bool HAS_BM>
__device__ __forceinline__ void wmma_store(float* __restrict__ C, int ldc,
                                           int m0, int n0, const v8f& acc,
                                           const float* __restrict__ bias_n,
                                           const float* __restrict__ bias_m) {
  const int lane = threadIdx.x & 31;
  const int half = lane >> 4;
  const int n    = lane & 15;
#pragma unroll
  for (int v = 0; v < 8; ++v) {
    const int m = m0 + half * 8 + v;
    float val = acc[v];
    if (HAS_BN) val += bias_n[n0 + n];   // compile-time dead when !HAS_BN
    if (HAS_BM) val += bias_m[m];        // compile-time dead when !HAS_BM
    C[(size_t)m * ldc + n0 + n] = val;
  }
}

// ---------------------------------------------------------------------------
// One-time: wenc[b][a][l] = sum_e W_enc[a][e] * enc_out[b][e][l] + b_enc[a]
// Per batch: C[ATTN x L] = W_enc[ATTN x ENC] * enc_out_b[ENC x L]
// Wave tile: 16 x 64.  Tiles: (ATTN/16) x (L/64) = 32 x 8 = 256 waves/batch.
// ---------------------------------------------------------------------------
__global__ void wenc_gemm_kernel(const float* __restrict__ W_enc,
                                 const float* __restrict__ enc_out,
                                 const float* __restrict__ b_enc,
                                 float* __restrict__ wenc) {
  const int b    = blockIdx.y;
  const int wave = threadIdx.x >> 5;
  const int tile = blockIdx.x * (blockDim.x >> 5) + wave;  // 256 wave-tiles
  const int TN   = LSEQ / 64;                              // 8 N-groups
  const int tm = tile / TN, tn = tile % TN;
  v8f acc[4] = {};
  wmma_accum_nt<0, 4>(W_enc, ENC, enc_out + (size_t)b * ENC * LSEQ, LSEQ,
                      ENC, tm * 16, tn * 64, acc);
  float* Cb = wenc + (size_t)b * ATTN * LSEQ;
#pragma unroll
  for (int j = 0; j < 4; ++j)
    wmma_store<false, true>(Cb, LSEQ, tm * 16, tn * 64 + j * 16, acc[j],
                            nullptr, b_enc);
}

// wdec[b][a] = h[b] . W_dec[a] + b_dec[a]   (C: [BS x ATTN], K = DEC)
// Tiles: (BS/16) x (ATTN/64) = 8 x 8 = 64 waves -> 8 blocks.
__global__ void wdec_gemm_kernel(const float* __restrict__ h,
                                 const float* __restrict__ W_dec,
                                 const float* __restrict__ b_dec,
                                 float* __restrict__ wdec) {
  const int wave = threadIdx.x >> 5;
  const int tile = blockIdx.x * (blockDim.x >> 5) + wave;
  const int TN = ATTN / 64;
  const int tm = tile / TN, tn = tile % TN;
  v8f acc[4] = {};
  wmma_accum_nt<1, 4>(h, DEC, W_dec, DEC, DEC, tm * 16, tn * 64, acc);
#pragma unroll
  for (int j = 0; j < 4; ++j)
    wmma_store<true, false>(wdec, ATTN, tm * 16, tn * 64 + j * 16, acc[j],
                            b_dec, nullptr);
}

// gates[b][j] = rnn_x[b].W_ih[j] + h[b].W_hh[j] + bias_g[j]   (N = 4*DEC)
// Tiles: (BS/16) x (4096/64) = 8 x 64 = 512 waves -> 64 blocks.
__global__ void gates_gemm_kernel(const float* __restrict__ rnn_x,
                                  const float* __restrict__ W_ih,
                                  const float* __restrict__ h,
                                  const float* __restrict__ W_hh,
                                  const float* __restrict__ bias_g,
                                  float* __restrict__ gates) {
  const int wave = threadIdx.x >> 5;
  const int tile = blockIdx.x * (blockDim.x >> 5) + wave;
  const int TN = (4 * DEC) / 64;
  const int tm = tile / TN, tn = tile % TN;
  v8f acc[4] = {};
  wmma_accum_nt<1, 4>(rnn_x, EMB + ENC, W_ih, EMB + ENC, EMB + ENC,
                      tm * 16, tn * 64, acc);
  wmma_accum_nt<1, 4>(h, DEC, W_hh, DEC, DEC, tm * 16, tn * 64, acc);
#pragma unroll
  for (int j = 0; j < 4; ++j)
    wmma_store<true, false>(gates, 4 * DEC, tm * 16, tn * 64 + j * 16, acc[j],
                            bias_g, nullptr);
}

// logits[b][v] = h[b] . W_out[v] + b_out[v]
// Tiles: (BS/16) x (VOCAB/64) = 8 x 2 = 16 waves -> 2 blocks.
__global__ void out_gemm_kernel(const float* __restrict__ h,
                                const float* __restrict__ W_out,
                                const float* __restrict__ b_out,
                                float* __restrict__ logits) {
  const int wave = threadIdx.x >> 5;
  const int tile = blockIdx.x * (blockDim.x >> 5) + wave;
  const int TN = VOCAB / 64;
  const int tm = tile / TN, tn = tile % TN;
  v8f acc[4] = {};
  wmma_accum_nt<1, 4>(h, DEC, W_out, DEC, DEC, tm * 16, tn * 64, acc);
#pragma unroll
  for (int j = 0; j < 4; ++j)
    wmma_store<true, false>(logits, VOCAB, tm * 16, tn * 64 + j * 16, acc[j],
                            b_out, nullptr);
}

// ---------------------------------------------------------------------------
// scores[b][l] = sum_a tanh(wenc[b][a][l] + wdec[b][a]) * v_e[a] + b_e
// One block per batch; each thread owns l and l+256 (fully coalesced stream
// of the 134 MB wenc tensor; wdec/v_e staged once in LDS).
// ---------------------------------------------------------------------------
__global__ void scores_kernel(const float* __restrict__ wenc,
                              const float* __restrict__ wdec,
                              const float* __restrict__ v_e,
                              const float* __restrict__ b_e,
                              float* __restrict__ scores) {
  __shared__ float s_wd[ATTN];
  __shared__ float s_ve[ATTN];
  const int b = blockIdx.x, tid = threadIdx.x;
  for (int i = tid; i < ATTN; i += 256) {
    s_wd[i] = wdec[(size_t)b * ATTN + i];
    s_ve[i] = v_e[i];
  }
  __syncthreads();
  const float* base = wenc + (size_t)b * ATTN * LSEQ;
  float s0 = 0.0f, s1 = 0.0f;
#pragma unroll 4
  for (int a = 0; a < ATTN; ++a) {
    const float w = s_wd[a], ve = s_ve[a];
    const float* row = base + (size_t)a * LSEQ;
    s0 += tanhf(row[tid] + w) * ve;
    s1 += tanhf(row[tid + 256] + w) * ve;
  }
  const float be = b_e[0];
  scores[(size_t)b * LSEQ + tid]       = s0 + be;
  scores[(size_t)b * LSEQ + tid + 256] = s1 + be;
}

// softmax over L; writes alphas (ws) and attn output slice for step t
__global__ void softmax_attn_kernel(const float* __restrict__ scores,
                                    float* __restrict__ alphas,
                                    float* __restrict__ attn_out, int t) {
  __shared__ float red[256];
  const int b = blockIdx.x, tid = threadIdx.x;
  const float* s = scores + (size_t)b * LSEQ;
  float x0 = s[tid], x1 = s[tid + 256];
  red[tid] = fmaxf(x0, x1);
  __syncthreads();
  for (int st = 128; st; st >>= 1) {
    if (tid < st) red[tid] = fmaxf(red[tid], red[tid + st]);
    __syncthreads();
  }
  const float mx = red[0];
  __syncthreads();
  const float e0 = __expf(x0 - mx), e1 = __expf(x1 - mx);
  red[tid] = e0 + e1;
  __syncthreads();
  for (int st = 128; st; st >>= 1) {
    if (tid < st) red[tid] += red[tid + st];
    __syncthreads();
  }
  const float inv = 1.0f / red[0];
  const float a0 = e0 * inv, a1 = e1 * inv;
  alphas[(size_t)b * LSEQ + tid] = a0;
  alphas[(size_t)b * LSEQ + tid + 256] = a1;
  float* ao = attn_out + ((size_t)b * TMAX + t) * LSEQ;
  ao[tid] = a0;
  ao[tid + 256] = a1;
}

// rnn_x[b][0:EMB] = emb_table[x_idx[b]]
__global__ void embed_kernel(const float* __restrict__ emb_table,
                             const int* __restrict__ x_idx,
                             float* __restrict__ rnn_x) {
  const int b = blockIdx.x, j = threadIdx.x;  // 256 threads == EMB
  rnn_x[(size_t)b * (EMB + ENC) + j] = emb_table[(size_t)x_idx[b] * EMB + j];
}

// rnn_x[b][EMB+e] = sum_l enc_out[b][e][l] * alphas[b][l]   (one wave per (b,e))
__global__ void context_kernel(const float* __restrict__ enc_out,
                               const float* __restrict__ alphas,
                               float* __restrict__ rnn_x) {
  const int wg   = blockIdx.x * (blockDim.x >> 5) + (threadIdx.x >> 5);
  const int b    = wg / ENC;
  const int e    = wg % ENC;
  const int lane = threadIdx.x & 31;
  const float* row = enc_out + ((size_t)b * ENC + e) * LSEQ;
  const float* al  = alphas + (size_t)b * LSEQ;
  float s = 0.0f;
  for (int l = lane; l < LSEQ; l += 32) s += row[l] * al[l];
#pragma unroll
  for (int off = 16; off; off >>= 1) s += __shfl_xor(s, off, 32);
  if (lane == 0) rnn_x[(size_t)b * (EMB + ENC) + EMB + e] = s;
}

__device__ __forceinline__ float sigm(float x) { return 1.0f / (1.0f + __expf(-x)); }

// LSTM cell elementwise (PyTorch gate order i,f,g,o)
__global__ void lstm_cell_kernel(const float* __restrict__ gates,
                                 float* __restrict__ h, float* __restrict__ c) {
  const int idx = blockIdx.x * blockDim.x + threadIdx.x;  // BS*DEC
  const int b = idx / DEC, d = idx % DEC;
  const float* g = gates + (size_t)b * 4 * DEC;
  const float i = sigm(g[d]);
  const float f = sigm(g[DEC + d]);
  const float gg = tanhf(g[2 * DEC + d]);
  const float o = sigm(g[3 * DEC + d]);
  const float cn = f * c[idx] + i * gg;
  c[idx] = cn;
  h[idx] = o * tanhf(cn);
}

// log_softmax + argmax over VOCAB; writes logp slice, preds, next x_idx
__global__ void head_kernel(const float* __restrict__ logits,
                            float* __restrict__ logp_out,
                            float* __restrict__ preds_out,
                            int* __restrict__ x_idx, int t) {
  __shared__ float sm[VOCAB];
  __shared__ int si[VOCAB];
  const int b = blockIdx.x, v = threadIdx.x;  // 128 threads
  const float x = logits[(size_t)b * VOCAB + v];
  sm[v] = x;
  __syncthreads();
  for (int s = 64; s; s >>= 1) {
    if (v < s) sm[v] = fmaxf(sm[v], sm[v + s]);
    __syncthreads();
  }
  const float mx = sm[0];
  __syncthreads();
  sm[v] = __expf(x - mx);
  __syncthreads();
  for (int s = 64; s; s >>= 1) {
    if (v < s) sm[v] += sm[v + s];
    __syncthreads();
  }
  const float lse = __logf(sm[0]) + mx;
  const float lp = x - lse;
  logp_out[((size_t)b * TMAX + t) * VOCAB + v] = lp;
  __syncthreads();
  sm[v] = lp;
  si[v] = v;
  __syncthreads();
  for (int s = 64; s; s >>= 1) {
    if (v < s && sm[v + s] > sm[v]) { sm[v] = sm[v + s]; si[v] = si[v + s]; }
    __syncthreads();
  }
  if (v == 0) {
    x_idx[b] = si[0];
    preds_out[(size_t)b * TMAX + t] = (float)si[0];
  }
}

// init h=c=0, x_idx=SOS, bias_g = b_ih + b_hh
__global__ void setup_kernel(float* __restrict__ h, float* __restrict__ c,
                             int* __restrict__ x_idx,
                             const float* __restrict__ b_ih,
                             const float* __restrict__ b_hh,
                             float* __restrict__ bias_g) {
  const int idx = blockIdx.x * blockDim.x + threadIdx.x;
  if (idx < BS * DEC) { h[idx] = 0.0f; c[idx] = 0.0f; }
  if (idx < 4 * DEC) bias_g[idx] = b_ih[idx] + b_hh[idx];
  if (idx < BS) x_idx[idx] = 0;  // SOS
}

extern "C" void kernel_launch(void* const* d_in, const int* in_sizes, int n_in,
                              void* d_out, int out_size, void* d_ws, size_t ws_size,
                              hipStream_t stream) {
  const float* enc_out   = (const float*)d_in[0];
  const float* W_enc     = (const float*)d_in[1];
  const float* b_enc     = (const float*)d_in[2];
  const float* W_dec     = (const float*)d_in[3];
  const float* b_dec     = (const float*)d_in[4];
  const float* v_e       = (const float*)d_in[5];
  const float* b_e       = (const float*)d_in[6];
  const float* emb_table = (const float*)d_in[7];
  const float* W_ih      = (const float*)d_in[8];
  const float* W_hh      = (const float*)d_in[9];
  const float* b_ih      = (const float*)d_in[10];
  const float* b_hh      = (const float*)d_in[11];
  const float* W_out     = (const float*)d_in[12];
  const float* b_out     = (const float*)d_in[13];

  // workspace layout (floats)
  float* ws = (float*)d_ws;
  size_t off = 0;
  float* wenc   = ws + off; off += (size_t)BS * ATTN * LSEQ;  // 134 MB
  float* h      = ws + off; off += (size_t)BS * DEC;
  float* c      = ws + off; off += (size_t)BS * DEC;
  float* wdec   = ws + off; off += (size_t)BS * ATTN;
  float* scores = ws + off; off += (size_t)BS * LSEQ;
  float* alphas = ws + off; off += (size_t)BS * LSEQ;
  float* rnn_x  = ws + off; off += (size_t)BS * (EMB + ENC);
  float* gates  = ws + off; off += (size_t)BS * 4 * DEC;
  float* logits = ws + off; off += (size_t)BS * VOCAB;
  float* bias_g = ws + off; off += (size_t)4 * DEC;
  int*   x_idx  = (int*)(ws + off); off += BS;

  // output layout: logp [BS,T,V] | preds [BS,T] | attn [BS,T,L]
  float* logp_out  = (float*)d_out;
  float* preds_out = logp_out + (size_t)BS * TMAX * VOCAB;
  float* attn_out  = preds_out + (size_t)BS * TMAX;

  setup_kernel<<<dim3((BS * DEC + 255) / 256), dim3(256), 0, stream>>>(
      h, c, x_idx, b_ih, b_hh, bias_g);

  // one-time weighted_enc: 256 wave-tiles/batch, 8 waves/block
  wenc_gemm_kernel<<<dim3(32, BS), dim3(256), 0, stream>>>(W_enc, enc_out, b_enc, wenc);

  for (int t = 0; t < TMAX; ++t) {
    // wdec = h @ W_dec^T + b_dec : 64 wave-tiles
    wdec_gemm_kernel<<<dim3(8), dim3(256), 0, stream>>>(h, W_dec, b_dec, wdec);
    // attention energies (streams wenc)
    scores_kernel<<<dim3(BS), dim3(256), 0, stream>>>(wenc, wdec, v_e, b_e, scores);
    // softmax + attn output
    softmax_attn_kernel<<<dim3(BS), dim3(256), 0, stream>>>(scores, alphas, attn_out, t);
    // embedding of previous token
    embed_kernel<<<dim3(BS), dim3(EMB), 0, stream>>>(emb_table, x_idx, rnn_x);
    // context = enc_out @ alphas (one wave per (b,e))
    context_kernel<<<dim3(BS * ENC / 8), dim3(256), 0, stream>>>(enc_out, alphas, rnn_x);
    // LSTM gates: 512 wave-tiles
    gates_gemm_kernel<<<dim3(64), dim3(256), 0, stream>>>(
        rnn_x, W_ih, h, W_hh, bias_g, gates);
    // cell update
    lstm_cell_kernel<<<dim3(BS * DEC / 256), dim3(256), 0, stream>>>(gates, h, c);
    // output projection: 16 wave-tiles
    out_gemm_kernel<<<dim3(2), dim3(256), 0, stream>>>(h, W_out, b_out, logits);
    // log_softmax + argmax + next token
    head_kernel<<<dim3(BS), dim3(VOCAB), 0, stream>>>(
        logits, logp_out, preds_out, x_idx, t);
  }
}